// TestEmbedding_22076131901759
// MI455X (gfx1250) — compile-verified
//
#include <hip/hip_runtime.h>
#include <stdint.h>

// ---------------------------------------------------------------------------
// Multi-level dense-grid trilinear embedding for MI455X (gfx1250, wave32).
// Gather-bound: all 9 grids (~66 MB) are L2-resident (192 MB L2). CDNA5 paths
// used: Tensor Data Mover (grid0 -> LDS, TENSORcnt), async global->LDS B128
// coordinate staging (ASYNCcnt), software-pipelined global_load_b64 gathers,
// coalesced B128 writeback via LDS transpose.
// ---------------------------------------------------------------------------

typedef unsigned int u32x4 __attribute__((ext_vector_type(4)));
typedef int          i32x4 __attribute__((ext_vector_type(4)));
typedef int          i32x8 __attribute__((ext_vector_type(8)));

struct GridPtrs { const float* p[9]; };

static __device__ inline float2 lerp2(float2 a, float2 b, float t) {
    float2 r;
    r.x = fmaf(t, b.x - a.x, a.x);
    r.y = fmaf(t, b.y - a.y, a.y);
    return r;
}

// Split trilinear lookup: issue() computes fracs + fires 8x global_load_b64;
// finish() does the z->y->x lerps. Lets the caller software-pipeline levels
// so ~4 levels of gathers are in flight per wave (L2-hit latency hiding).
template <int G>
struct Gather {
    float tx, ty, tz;
    float2 v[8];

    __device__ inline void issue(const float* __restrict__ g,
                                 float x, float y, float z) {
        constexpr float S = (float)(G - 1);
        float cx = x * S, cy = y * S, cz = z * S;
        float fx = floorf(cx), fy = floorf(cy), fz = floorf(cz);
        int ix = (int)fx, iy = (int)fy, iz = (int)fz;
        ix = ix < 0 ? 0 : (ix > G - 2 ? G - 2 : ix);
        iy = iy < 0 ? 0 : (iy > G - 2 ? G - 2 : iy);
        iz = iz < 0 ? 0 : (iz > G - 2 ? G - 2 : iz);
        tx = cx - (float)ix; ty = cy - (float)iy; tz = cz - (float)iz;

        const float* b = g + 2 * (((ix * G) + iy) * G + iz);
        constexpr int YS = 2 * G;       // +1 in y
        constexpr int XS = 2 * G * G;   // +1 in x
        v[0] = *(const float2*)(b);
        v[1] = *(const float2*)(b + 2);
        v[2] = *(const float2*)(b + YS);
        v[3] = *(const float2*)(b + YS + 2);
        v[4] = *(const float2*)(b + XS);
        v[5] = *(const float2*)(b + XS + 2);
        v[6] = *(const float2*)(b + XS + YS);
        v[7] = *(const float2*)(b + XS + YS + 2);
    }

    __device__ inline float2 finish() const {
        float2 a0 = lerp2(v[0], v[1], tz);
        float2 a1 = lerp2(v[2], v[3], tz);
        float2 a2 = lerp2(v[4], v[5], tz);
        float2 a3 = lerp2(v[6], v[7], tz);
        float2 b0 = lerp2(a0, a1, ty);
        float2 b1 = lerp2(a2, a3, ty);
        return lerp2(b0, b1, tx);
    }
};

__global__ __launch_bounds__(256)
void hashgrid_encode_kernel(const float* __restrict__ in, GridPtrs gp,
                            float* __restrict__ out, int N) {
    __shared__ __align__(16) float s_in[256 * 3];    // staged coords
    __shared__ __align__(16) float s_out[256 * 21];  // staged outputs
    __shared__ __align__(16) float s_g0[256];        // grid0: 250 f32 via TDM

    const int tid = threadIdx.x;
    const int n0  = blockIdx.x << 8;
    int nvalid = N - n0;
    if (nvalid > 256) nvalid = 256;

    // ---- Tensor Data Mover: DMA grid0 (5*5*5*2 f32 = 1000 B) into LDS ----
    // Issued by wave 0 only (TENSORcnt is per-wave). Descriptor per CDNA5 ISA
    // ch.8: 1-D tensor of 250 4-byte elements, 1-D tile of 250.
    if (tid < 32) {
        unsigned long long ga = (unsigned long long)(uintptr_t)gp.p[0];
        unsigned int ldsa = (unsigned int)(uintptr_t)&s_g0[0]; // low 32b of flat = LDS offset
        u32x4 g0;
        g0.x = 1u;                                   // count=1 (valid user D#)
        g0.y = ldsa;                                 // lds_addr (bytes)
        g0.z = (unsigned int)ga;                     // global_addr[31:0]
        g0.w = (unsigned int)((ga >> 32) & 0x01FFFFFFull) | 0x80000000u; // [56:32] | type=2
        i32x8 g1;
        g1[0] = 0x00020000;                          // data_size=2 (4B); no multicast/pad
        g1[1] = (int)(250u << 16);                   // tensor_dim0 = 250
        g1[2] = (int)(1u << 16);                     // tensor_dim1 = 1
        g1[3] = (int)(250u << 16);                   // tile_dim0 = 250
        g1[4] = 0;                                   // tile_dim1 = tile_dim2 = 0 (1-D tile)
        g1[5] = 250;                                 // tensor_dim0_stride = 250
        g1[6] = 0;
        g1[7] = 0;
        i32x4 gz = {0, 0, 0, 0};
#if defined(__has_include)
#if __has_include(<hip/amd_detail/amd_gfx1250_TDM.h>)
        i32x8 gz8 = {0, 0, 0, 0, 0, 0, 0, 0};
        __builtin_amdgcn_tensor_load_to_lds(g0, g1, gz, gz, gz8, 0);   // 6-arg toolchain
#else
        __builtin_amdgcn_tensor_load_to_lds(g0, g1, gz, gz, 0);        // 5-arg ROCm 7.2
#endif
#else
        __builtin_amdgcn_tensor_load_to_lds(g0, g1, gz, gz, 0);
#endif
        __builtin_amdgcn_s_wait_tensorcnt(0);
    }

    // ---- stage this block's coordinates into LDS ----
    const float* src = in + (size_t)n0 * 3;
    if (nvalid == 256) {
        // Async DMA: 768 floats = 192 lanes x B128, ASYNCcnt-tracked.
        if (tid < 192) {
            unsigned int lds_addr = (unsigned int)(uintptr_t)&s_in[0] + (unsigned int)tid * 16u;
            const char* gaddr = (const char*)src + (size_t)tid * 16u;
            asm volatile("global_load_async_to_lds_b128 %0, %1, off"
                         :: "v"(lds_addr), "v"(gaddr) : "memory");
        }
        asm volatile("s_wait_asynccnt 0x0" ::: "memory");
    } else {
        int remain = nvalid * 3;
        for (int k = tid; k < remain; k += 256) s_in[k] = src[k];
    }

    // prefetch next block's coordinates toward L2 (global_prefetch_b8)
    if (n0 + 256 + tid < N) {
        __builtin_prefetch(in + (size_t)(n0 + 256) * 3 + tid, 0, 3);
    }

    __syncthreads();

    // ---- per-point compute: 9 trilinear lookups, ~4-level pipelined ----
    if (tid < nvalid) {
        float x = s_in[tid * 3 + 0];
        float y = s_in[tid * 3 + 1];
        float z = s_in[tid * 3 + 2];
        float* po = &s_out[tid * 21];       // stride 21 coprime w/ 64 banks
        po[0] = x; po[1] = y; po[2] = z;

        Gather<61>  q1; Gather<81>  q2; Gather<65> q3; Gather<85> q4;
        Gather<141> q5; Gather<65>  q6; Gather<85> q7; Gather<141> q8;

        q1.issue(gp.p[1], x, y, z);
        q2.issue(gp.p[2], x, y, z);
        q3.issue(gp.p[3], x, y, z);
        q4.issue(gp.p[4], x, y, z);

        // level 0 from TDM-staged LDS copy (ds_load_b64, low latency)
        {
            Gather<5> q0;
            q0.issue(s_g0, x, y, z);
            float2 r = q0.finish();
            po[3] = r.x; po[4] = r.y;
        }

        q5.issue(gp.p[5], x, y, z);
        { float2 r = q1.finish(); po[5]  = r.x; po[6]  = r.y; }
        q6.issue(gp.p[6], x, y, z);
        { float2 r = q2.finish(); po[7]  = r.x; po[8]  = r.y; }
        q7.issue(gp.p[7], x, y, z);
        { float2 r = q3.finish(); po[9]  = r.x; po[10] = r.y; }
        q8.issue(gp.p[8], x, y, z);
        { float2 r = q4.finish(); po[11] = r.x; po[12] = r.y; }
        { float2 r = q5.finish(); po[13] = r.x; po[14] = r.y; }
        { float2 r = q6.finish(); po[15] = r.x; po[16] = r.y; }
        { float2 r = q7.finish(); po[17] = r.x; po[18] = r.y; }
        { float2 r = q8.finish(); po[19] = r.x; po[20] = r.y; }
    }
    __syncthreads();

    // ---- coalesced writeback: 256*21 floats contiguous per block ----
    const size_t obase = (size_t)n0 * 21;
    if (nvalid == 256) {
        // 5376 floats = 1344 float4; block base = n0*84 B, multiple of 16. B128 stores.
        float4* o4 = (float4*)(out + obase);
        const float4* s4 = (const float4*)s_out;
#pragma unroll
        for (int k = 0; k < 6; ++k) {
            int idx = tid + (k << 8);
            if (idx < 1344) o4[idx] = s4[idx];
        }
    } else {
        int total = nvalid * 21;
        for (int k = tid; k < total; k += 256) out[obase + k] = s_out[k];
    }
}

extern "C" void kernel_launch(void* const* d_in, const int* in_sizes, int n_in,
                              void* d_out, int out_size, void* d_ws, size_t ws_size,
                              hipStream_t stream) {
    const float* inputs = (const float*)d_in[0];
    GridPtrs gp;
    for (int i = 0; i < 9; ++i) gp.p[i] = (const float*)d_in[1 + i];
    float* out = (float*)d_out;

    const int N = in_sizes[0] / 3;           // [N,3] coords
    const int blocks = (N + 255) / 256;
    hipLaunchKernelGGL(hashgrid_encode_kernel, dim3(blocks), dim3(256), 0, stream,
                       inputs, gp, out, N);
}